// Render_58153857188549
// MI455X (gfx1250) — compile-verified
//
#include <hip/hip_runtime.h>

// Fixed problem shape from the reference.
#define H_   256
#define W_   256
#define B_   8
#define N_   65536            // 2^16 -> batch = p >> 16
#define HW_  (H_ * W_)
#define NPTS (B_ * N_)        // 524288
#define EPS_ 1e-5f
#define BIG_ 1e10f

typedef __attribute__((ext_vector_type(2))) float v2f;
typedef __attribute__((ext_vector_type(8))) float v8f;

// v_rcp_f32 (~1 ulp) + one Newton-Raphson step -> ~0.5 ulp at 3 VALU ops,
// instead of the ~10-op div_scale/div_fixup sequence in the hot scatter loop.
__device__ __forceinline__ float fast_rcp(float d) {
    float r = __builtin_amdgcn_rcpf(d);
    return r * (2.0f - d * r);
}

// ---------------------------------------------------------------------------
// K0: initialize z-buffer (workspace) to BIG and output images to 0.
// Must run every launch: harness poisons d_out/d_ws only once.
// ---------------------------------------------------------------------------
__global__ void rend_init_kernel(float* __restrict__ zbuf,
                                 float* __restrict__ dimg,
                                 float* __restrict__ wimg) {
    int i = blockIdx.x * blockDim.x + threadIdx.x;   // grid == B*H*W exactly
    zbuf[i] = BIG_;
    dimg[i] = 0.0f;
    wimg[i] = 0.0f;
}

// ---------------------------------------------------------------------------
// K1: scatter-min depth over 5x5 patch. z in [1,2) and BIG are positive, so
// uint compare of IEEE bits == float compare -> hardware global_atomic_min_u32.
// ---------------------------------------------------------------------------
__global__ void rend_zmin_kernel(const float* __restrict__ pts,
                                 unsigned*    __restrict__ zbuf) {
    int p = blockIdx.x * blockDim.x + threadIdx.x;   // grid == NPTS exactly
    float x = pts[3 * p + 0];
    float y = pts[3 * p + 1];
    float z = pts[3 * p + 2];
    int px = (int)rintf(x);                          // RNE == jnp.round
    int py = (int)rintf(y);
    unsigned zi = __float_as_uint(z);
    unsigned* base = zbuf + (p >> 16) * HW_;
#pragma unroll
    for (int di = -2; di <= 2; ++di) {
        int ii = py + di;
        if (ii < 0 || ii >= H_) continue;
#pragma unroll
        for (int dj = -2; dj <= 2; ++dj) {
            int jj = px + dj;
            if (jj < 0 || jj >= W_) continue;
            atomicMin(base + ii * W_ + jj, zi);
        }
    }
}

// ---------------------------------------------------------------------------
// K2: visibility test + weighted splat.
// Each wave32 owns 16 points. d2(m,n) for 16 points x 64 patch slots (49 valid)
// is computed with four V_WMMA_F32_16X16X4_F32 via the rank-4 decomposition:
//   d2 = 1*(dj^2+di^2) + fx*(-2dj) + fy*(-2di) + (fx^2+fy^2)*1
// A (16x4) layout per ISA: lanes 0-15 hold K0,K1 in v[0],v[1]; lanes 16-31 hold
// K2,K3. C/D layout: lane l, VGPR r -> M = r + 8*(l>=16), N = l&15.
// ---------------------------------------------------------------------------
__global__ void rend_splat_kernel(const float* __restrict__ pts,
                                  const float* __restrict__ zbuf,
                                  const float* __restrict__ thr_p,
                                  float* __restrict__ dimg,
                                  float* __restrict__ wimg,
                                  float* __restrict__ visout) {
    const float thr = *thr_p;
    const int lane = threadIdx.x & 31;
    const int wid  = threadIdx.x >> 5;
    const int ls   = lane & 15;
    const int pbase = (blockIdx.x * 8 + wid) * 16;   // 8 waves/block, 16 pts/wave

    // ---- lanes 0-15 load one point each (shapes exact, no tail) ----
    float x = 0.f, y = 0.f, z = 0.f, fx = 0.f, fy = 0.f, s = 0.f;
    int px = 0, py = 0, ibase = 0, vis = 0;
    if (lane < 16) {
        int p = pbase + lane;
        x = pts[3 * p + 0];
        y = pts[3 * p + 1];
        z = pts[3 * p + 2];
        px = (int)rintf(x);
        py = (int)rintf(y);
        ibase = (p >> 16) * HW_;
        bool in_img = (px >= 0) & (px < W_) & (py >= 0) & (py < H_);
        float zmin = in_img ? zbuf[ibase + py * W_ + px] : BIG_;
        vis = (in_img && (z <= zmin + thr)) ? 1 : 0;
        visout[p] = vis ? 1.0f : 0.0f;               // bool output as 0/1 float
        fx = x - (float)px;
        fy = y - (float)py;
        s  = fx * fx + fy * fy;
    }

    // ---- build A matrix (fully converged from here: WMMA needs EXEC==~0) ----
    float fxm = __shfl(fx, ls, 32);
    float fym = __shfl(fy, ls, 32);
    float sm  = __shfl(s,  ls, 32);
    v2f a;
    a.x = (lane < 16) ? 1.0f : fym;                  // K0 | K2
    a.y = (lane < 16) ? fxm  : sm;                   // K1 | K3

    // ---- four 16x16x4 f32 WMMAs: d2 for patch slots nn = ls + 16*blk ----
    v8f D[4];
#pragma unroll
    for (int blk = 0; blk < 4; ++blk) {
        int nn = ls + blk * 16;                      // slots >= 49 masked later
        float fdi = (float)(nn / 7 - 3);
        float fdj = (float)(nn % 7 - 3);
        v2f bm;
        bm.x = (lane < 16) ? (fdj * fdj + fdi * fdi) : (-2.0f * fdi); // K0 | K2
        bm.y = (lane < 16) ? (-2.0f * fdj)           : 1.0f;          // K1 | K3
        v8f c = {};
        D[blk] = __builtin_amdgcn_wmma_f32_16x16x4_f32(
            /*neg_a=*/false, a, /*neg_b=*/false, bm,
            /*c_mod=*/(short)0, c, /*reuse_a=*/false, /*reuse_b=*/false);
    }

    // ---- scatter: lane handles point m = r + 8*(lane>=16), slot n = ls+16*blk
    const int half = lane >> 4;
#pragma unroll
    for (int r = 0; r < 8; ++r) {
        int   m    = r + 8 * half;
        int   pxm  = __shfl(px, m, 32);
        int   pym  = __shfl(py, m, 32);
        float zm   = __shfl(z,  m, 32);
        int   vism = __shfl(vis, m, 32);
        int   bsm  = __shfl(ibase, m, 32);
#pragma unroll
        for (int blk = 0; blk < 4; ++blk) {
            int nn = ls + blk * 16;
            if (vism && nn < 49) {
                int ii = pym + (nn / 7 - 3);
                int jj = pxm + (nn % 7 - 3);
                if (ii >= 0 && ii < H_ && jj >= 0 && jj < W_) {
                    float w  = fast_rcp(D[blk][r] + EPS_);
                    int addr = bsm + ii * W_ + jj;
                    unsafeAtomicAdd(&wimg[addr], w);        // global_atomic_add_f32
                    unsafeAtomicAdd(&dimg[addr], w * zm);
                }
            }
        }
    }
}

// ---------------------------------------------------------------------------
// d_out layout: depth[B*H*W] | weight[B*H*W] | is_visible[B*N]  (floats)
// d_ws: z-buffer, B*H*W floats (2 MB)
// ---------------------------------------------------------------------------
extern "C" void kernel_launch(void* const* d_in, const int* in_sizes, int n_in,
                              void* d_out, int out_size, void* d_ws, size_t ws_size,
                              hipStream_t stream) {
    const float* pts = (const float*)d_in[0];   // [B, N, 3]
    const float* thr = (const float*)d_in[1];   // scalar

    float* dimg   = (float*)d_out;
    float* wimg   = dimg + (size_t)B_ * HW_;
    float* visout = wimg + (size_t)B_ * HW_;
    float* zbuf   = (float*)d_ws;

    (void)in_sizes; (void)n_in; (void)out_size; (void)ws_size;

    rend_init_kernel<<<(B_ * HW_) / 256, 256, 0, stream>>>(zbuf, dimg, wimg);
    rend_zmin_kernel<<<NPTS / 256, 256, 0, stream>>>(pts, (unsigned*)zbuf);
    rend_splat_kernel<<<NPTS / 128, 256, 0, stream>>>(pts, zbuf, thr,
                                                      dimg, wimg, visout);
}